// BeamSearchAttentionDecoder_46866683134213
// MI455X (gfx1250) — compile-verified
//
#include <hip/hip_runtime.h>
#include <math.h>

// ---------------------------------------------------------------------------
// BeamSearchAttentionDecoder for MI455X (gfx1250, wave32, WMMA)
// H=512, S=1024, B=16, TOPK=4, BK=64, PREV=3
//
// Algebraic refactor: attention uses q = h@w1 against raw encoder stream;
// context = w2 @ (sum_s align_s * enc_s) + b2 (softmax weights sum to 1).
// Dense GEMMs use V_WMMA_F32_16X16X4_F32 (full fp32, one wave per 16x16 tile),
// K stepped by 8 with b128 loads feeding two WMMAs via K-slot permutation.
// ---------------------------------------------------------------------------

typedef float v2f __attribute__((ext_vector_type(2)));
typedef float v8f __attribute__((ext_vector_type(8)));

#define HH   512
#define SS   1024
#define BK   64
#define NEG_INF_F (-10000000000.0f)
#define INV_SQRT_H 0.04419417382415922f  // 1/sqrt(512)

// ---------------- WMMA fp32 GEMM: C[M,N] = A[M,K] @ op(B) (+bias) ----------
// TRANSB=0: B stored [K,N] row-major.  TRANSB=1: B stored [N,K] row-major.
// One wave computes one 16x16 tile; 8 waves (256 thr) per block.
// A 16x4 layout: lanes 0-15 -> M=lane, slot K pair {p0,p0+1}; lanes 16-31 hi half.
// K is processed in chunks of 8: wmma#1 carries logical K {k,k+1,k+4,k+5},
// wmma#2 carries {k+2,k+3,k+6,k+7}. The slot->k association matches between
// A and B, so the contraction is exact while each lane loads ONE float4
// (global_load_b128) per matrix per 8 K values.
template <int TRANSB>
__global__ void gemm_wmma_f32(const float* __restrict__ A, const float* __restrict__ B,
                              const float* __restrict__ bias, float* __restrict__ C,
                              int M, int N, int K)
{
    int lane = threadIdx.x & 31;
    int wave = threadIdx.x >> 5;
    int ntn  = N >> 4;
    int T    = (M >> 4) * ntn;
    int t    = blockIdx.x * 8 + wave;          // wave-uniform -> EXEC all-ones
    if (t >= T) return;
    int tm = (t / ntn) << 4;
    int tn = (t % ntn) << 4;
    int l15 = lane & 15;
    int hi  = lane >> 4;                        // 0 or 1
    int row = tm + l15;
    int col = tn + l15;

    v8f acc = {};
    const float4* A4 = (const float4*)(A + (size_t)row * K);

    if (TRANSB) {
        const float4* B4 = (const float4*)(B + (size_t)col * K);
        int nk8 = K >> 3;
        #pragma unroll 4
        for (int k8 = 0; k8 < nk8; k8++) {
            float4 av = A4[k8 * 2 + hi];
            float4 bv = B4[k8 * 2 + hi];
            v2f a0; a0.x = av.x; a0.y = av.y;
            v2f a1; a1.x = av.z; a1.y = av.w;
            v2f b0; b0.x = bv.x; b0.y = bv.y;
            v2f b1; b1.x = bv.z; b1.y = bv.w;
            acc = __builtin_amdgcn_wmma_f32_16x16x4_f32(false, a0, false, b0,
                                                        (short)0, acc, false, false);
            acc = __builtin_amdgcn_wmma_f32_16x16x4_f32(false, a1, false, b1,
                                                        (short)0, acc, false, false);
        }
    } else {
        #pragma unroll 4
        for (int k = 0; k < K; k += 8) {
            float4 av = A4[(k >> 2) + hi];
            int kb = k + 4 * hi;                // rows of B this lane carries
            v2f a0; a0.x = av.x; a0.y = av.y;
            v2f a1; a1.x = av.z; a1.y = av.w;
            v2f b0, b1;
            b0.x = B[(size_t)(kb + 0) * N + col];
            b0.y = B[(size_t)(kb + 1) * N + col];
            b1.x = B[(size_t)(kb + 2) * N + col];
            b1.y = B[(size_t)(kb + 3) * N + col];
            acc = __builtin_amdgcn_wmma_f32_16x16x4_f32(false, a0, false, b0,
                                                        (short)0, acc, false, false);
            acc = __builtin_amdgcn_wmma_f32_16x16x4_f32(false, a1, false, b1,
                                                        (short)0, acc, false, false);
        }
    }

    float bv = bias ? bias[col] : 0.0f;
    #pragma unroll
    for (int i = 0; i < 8; i++) {
        int r = tm + i + (hi ? 8 : 0);
        C[(size_t)r * N + col] = acc[i] + bv;
    }
}

// ---------------- GRU gates: h_new from gi (x@Wih^T+bih) and gh ------------
__global__ void gru_gate_kernel(const float* __restrict__ gi, const float* __restrict__ gh,
                                const float* __restrict__ hprev, float* __restrict__ hnew)
{
    int g = blockIdx.x * blockDim.x + threadIdx.x;   // 64*512
    int b = g >> 9, h = g & 511;
    const float* gib = gi + (size_t)b * 1536;
    const float* ghb = gh + (size_t)b * 1536;
    float r = 1.0f / (1.0f + __expf(-(gib[h]        + ghb[h])));
    float z = 1.0f / (1.0f + __expf(-(gib[512 + h]  + ghb[512 + h])));
    float n = tanhf(gib[1024 + h] + r * ghb[1024 + h]);
    hnew[g] = (1.0f - z) * n + z * hprev[g];
}

// ---------------- qb[b] = dot(h_new[b], b1) --------------------------------
__global__ void dot_hb1_kernel(const float* __restrict__ hnew, const float* __restrict__ b1,
                               float* __restrict__ qb)
{
    __shared__ float red[16];
    int b = blockIdx.x, t = threadIdx.x;             // 512 threads
    float v = hnew[(size_t)b * HH + t] * b1[t];
    for (int off = 16; off; off >>= 1) v += __shfl_xor(v, off);
    if ((t & 31) == 0) red[t >> 5] = v;
    __syncthreads();
    if (t < 32) {
        v = (t < 16) ? red[t] : 0.0f;
        for (int off = 16; off; off >>= 1) v += __shfl_xor(v, off);
        if (t == 0) qb[b] = v;
    }
}

// ---------------- streaming score pass: attn[b,s] = (q.enc+qb)/sqrtH + mask -
// grid (64, 8), block 512 (16 waves). One wave = one s per step, lane owns 16 h.
__global__ void attn_score_kernel(const float* __restrict__ enc, const float* __restrict__ q,
                                  const float* __restrict__ qb, const float* __restrict__ mask,
                                  float* __restrict__ attn)
{
    int b     = blockIdx.x;
    int sbase = blockIdx.y * 128;
    int lane  = threadIdx.x & 31;
    int wave  = threadIdx.x >> 5;                    // 0..15
    const float4* q4 = (const float4*)(q + (size_t)b * HH);
    float4 qv[4];
    #pragma unroll
    for (int j = 0; j < 4; j++) qv[j] = q4[lane * 4 + j];
    float qbb = qb[b];
    for (int i = 0; i < 8; i++) {
        int s = sbase + wave + i * 16;
        const float4* e4 = (const float4*)(enc + ((size_t)b * SS + s) * HH);
        if (s + 16 < SS)   // speculative prefetch of next row chunk (global_prefetch_b8)
            __builtin_prefetch(enc + ((size_t)b * SS + s + 16) * HH + lane * 16, 0, 1);
        float sum = 0.0f;
        #pragma unroll
        for (int j = 0; j < 4; j++) {
            float4 e = e4[lane * 4 + j];
            sum += e.x * qv[j].x + e.y * qv[j].y + e.z * qv[j].z + e.w * qv[j].w;
        }
        for (int off = 16; off; off >>= 1) sum += __shfl_xor(sum, off);
        if (lane == 0)
            attn[(size_t)b * SS + s] = (sum + qbb) * INV_SQRT_H + mask[(size_t)b * SS + s];
    }
}

// ---------------- softmax over S + top-4 (stable, ties -> smallest idx) ----
__global__ void softmax_topk_kernel(const float* __restrict__ attn, float* __restrict__ align,
                                    float* __restrict__ scores, int* __restrict__ sents)
{
    int b = blockIdx.x;
    int t = threadIdx.x;                             // 1024 threads = 32 waves
    int lid = t & 31, wid = t >> 5;
    __shared__ float sa[1024];
    __shared__ float red[32];
    __shared__ float rv[32];
    __shared__ int   ri[32];
    __shared__ float s_m, s_l;

    float a = attn[(size_t)b * SS + t];
    sa[t] = a;
    float m = a;
    for (int off = 16; off; off >>= 1) m = fmaxf(m, __shfl_xor(m, off));
    if (lid == 0) red[wid] = m;
    __syncthreads();
    if (wid == 0) {
        float v = red[lid];
        for (int off = 16; off; off >>= 1) v = fmaxf(v, __shfl_xor(v, off));
        if (lid == 0) s_m = v;
    }
    __syncthreads();
    float e = __expf(a - s_m);
    float l = e;
    for (int off = 16; off; off >>= 1) l += __shfl_xor(l, off);
    if (lid == 0) red[wid] = l;
    __syncthreads();
    if (wid == 0) {
        float v = red[lid];
        for (int off = 16; off; off >>= 1) v += __shfl_xor(v, off);
        if (lid == 0) s_l = v;
    }
    __syncthreads();
    align[(size_t)b * SS + t] = e / s_l;

    // top-4 of pre-softmax scores == top-4 of softmax (monotone)
    for (int kk = 0; kk < 4; kk++) {
        float v = sa[t]; int idx = t;
        for (int off = 16; off; off >>= 1) {
            float ov = __shfl_xor(v, off); int oi = __shfl_xor(idx, off);
            if (ov > v || (ov == v && oi < idx)) { v = ov; idx = oi; }
        }
        if (lid == 0) { rv[wid] = v; ri[wid] = idx; }
        __syncthreads();
        if (t == 0) {
            float bv = rv[0]; int bi = ri[0];
            for (int w = 1; w < 32; w++)
                if (rv[w] > bv || (rv[w] == bv && ri[w] < bi)) { bv = rv[w]; bi = ri[w]; }
            scores[b * 4 + kk] = __expf(bv - s_m) / s_l;
            sents[b * 4 + kk]  = bi;
            sa[bi] = -3.0e38f;
        }
        __syncthreads();
    }
}

// ---------------- ctx_raw[b,h] = sum_s align[b,s] * enc[b,s,h] -------------
__global__ void ctx_accum_kernel(const float* __restrict__ enc, const float* __restrict__ align,
                                 float* __restrict__ ctxr)
{
    int b = blockIdx.x, t = threadIdx.x;             // 512 threads
    __shared__ float al[1024];
    al[t]       = align[(size_t)b * SS + t];
    al[t + 512] = align[(size_t)b * SS + 512 + t];
    __syncthreads();
    float acc = 0.0f;
    const float* eb = enc + (size_t)b * SS * HH;
    #pragma unroll 4
    for (int s = 0; s < SS; s++)
        acc = fmaf(al[s], eb[(size_t)s * HH + t], acc);
    ctxr[(size_t)b * HH + t] = acc;
}

// ---------------- cat = [context, h_new] -----------------------------------
__global__ void concat_kernel(const float* __restrict__ ctx, const float* __restrict__ hnew,
                              float* __restrict__ cat)
{
    int g = blockIdx.x * blockDim.x + threadIdx.x;   // 64*1024
    int b = g >> 10, h = g & 1023;
    cat[g] = (h < 512) ? ctx[(size_t)b * HH + h] : hnew[(size_t)b * HH + h - 512];
}

// ---------------- beam selection (stable 4-smallest of 16) -----------------
__global__ void select_kernel(const float* __restrict__ scores, const int* __restrict__ sents,
                              const float* __restrict__ ev, const int* __restrict__ evidx,
                              int* __restrict__ sarr, int* __restrict__ sel,
                              float* __restrict__ out_scores, int* __restrict__ out_idx)
{
    int rb = threadIdx.x;
    if (rb >= 16) return;
    float flat[16]; bool used[16];
    for (int j = 0; j < 16; j++) {
        int gb = rb * 4 + (j >> 2);
        int r  = j & 3;
        flat[j] = -logf(scores[gb * 4 + r]) + ev[gb];
        used[j] = false;
    }
    for (int m = 0; m < 4; m++) {
        int best = 0; float bv = 3.0e38f;
        for (int j = 0; j < 16; j++)
            if (!used[j] && flat[j] < bv) { bv = flat[j]; best = j; }
        used[best] = true;
        int o  = rb * 4 + m;
        int sg = rb * 4 + (best >> 2);
        int r  = best & 3;
        sarr[o] = sg;
        int ssent = sents[sg * 4 + r];
        sel[o] = ssent;
        out_scores[o] = bv;
        out_idx[o * 4 + 0] = evidx[sg * 3 + 0];
        out_idx[o * 4 + 1] = evidx[sg * 3 + 1];
        out_idx[o * 4 + 2] = evidx[sg * 3 + 2];
        out_idx[o * 4 + 3] = ssent;
    }
}

// ---------------- gathers ---------------------------------------------------
__global__ void gather_rh_kernel(const float* __restrict__ res, const float* __restrict__ hnew,
                                 const int* __restrict__ sarr,
                                 float* __restrict__ out_res, float* __restrict__ out_hid)
{
    int g = blockIdx.x * blockDim.x + threadIdx.x;   // 32768
    int o = g >> 9, h = g & 511;
    int s = sarr[o];
    out_res[g] = res[(size_t)s * HH + h];
    out_hid[g] = hnew[(size_t)s * HH + h];
}

__global__ void gather_attn_kernel(const float* __restrict__ prev, const float* __restrict__ attn,
                                   const int* __restrict__ sarr, float* __restrict__ out_attn)
{
    int g = blockIdx.x * blockDim.x + threadIdx.x;   // 4*64*1024
    int p = g >> 16;
    int o = (g >> 10) & 63;
    int s = g & 1023;
    int sb = sarr[o];
    out_attn[g] = (p < 3) ? prev[((size_t)p * BK + sb) * SS + s]
                          : attn[(size_t)sb * SS + s];
}

__global__ void gather_mask_kernel(const float* __restrict__ mask, const int* __restrict__ sarr,
                                   const int* __restrict__ sel, float* __restrict__ out_mask)
{
    int g = blockIdx.x * blockDim.x + threadIdx.x;   // 65536
    int o = g >> 10, s = g & 1023;
    float v = mask[(size_t)sarr[o] * SS + s];
    if (s == sel[o]) v = NEG_INF_F;
    out_mask[g] = v;
}

// ---------------------------------------------------------------------------
extern "C" void kernel_launch(void* const* d_in, const int* in_sizes, int n_in,
                              void* d_out, int out_size, void* d_ws, size_t ws_size,
                              hipStream_t stream)
{
    const float* last_hidden = (const float*)d_in[0];   // (1,64,512)
    const float* dec_in      = (const float*)d_in[1];   // (64,1,512)
    const float* enc         = (const float*)d_in[2];   // (64,1024,512)
    const float* attn_prev   = (const float*)d_in[3];   // (3,64,1,1024)
    const float* mask_in     = (const float*)d_in[4];   // (64,1,1024)
    const float* ev_sc       = (const float*)d_in[5];   // (64,)
    const int*   ev_idx      = (const int*)  d_in[6];   // (64,3) int32
    const float* w1   = (const float*)d_in[7];
    const float* b1   = (const float*)d_in[8];
    const float* w2   = (const float*)d_in[9];
    const float* b2   = (const float*)d_in[10];
    const float* w3   = (const float*)d_in[11];
    const float* b3   = (const float*)d_in[12];
    const float* w_ih = (const float*)d_in[13];
    const float* w_hh = (const float*)d_in[14];
    const float* b_ih = (const float*)d_in[15];
    const float* b_hh = (const float*)d_in[16];

    // workspace layout (floats)
    float* ws = (float*)d_ws;
    size_t o = 0;
    float* GI    = ws + o; o += (size_t)BK * 1536;
    float* GH    = ws + o; o += (size_t)BK * 1536;
    float* HNEW  = ws + o; o += (size_t)BK * HH;
    float* Q     = ws + o; o += (size_t)BK * HH;
    float* QB    = ws + o; o += BK;
    float* ATTN  = ws + o; o += (size_t)BK * SS;
    float* ALIGN = ws + o; o += (size_t)BK * SS;
    float* SC    = ws + o; o += BK * 4;
    int*   SE    = (int*)(ws + o); o += BK * 4;
    float* CTXR  = ws + o; o += (size_t)BK * HH;
    float* CTX   = ws + o; o += (size_t)BK * HH;
    float* CAT   = ws + o; o += (size_t)BK * 1024;
    float* RES   = ws + o; o += (size_t)BK * HH;
    int*   SARR  = (int*)(ws + o); o += BK;
    int*   SEL   = (int*)(ws + o); o += BK;

    // output layout (flat, return order)
    float* out        = (float*)d_out;
    float* out_result = out;                        // 64*512
    float* out_hidden = out + 32768;                // 64*512
    int*   out_idx    = (int*)(out + 65536);        // 64*4 int32 (raw bits)
    float* out_attn   = out + 65792;                // 4*64*1024
    float* out_mask   = out + 327936;               // 64*1024
    float* out_scores = out + 393472;               // 64

    // --- GRU gates: gi = x @ w_ih^T + b_ih ; gh = h @ w_hh^T + b_hh
    int T_g = (BK / 16) * (1536 / 16);              // 384 tiles
    gemm_wmma_f32<1><<<(T_g + 7) / 8, 256, 0, stream>>>(dec_in,      w_ih, b_ih, GI, BK, 1536, HH);
    gemm_wmma_f32<1><<<(T_g + 7) / 8, 256, 0, stream>>>(last_hidden, w_hh, b_hh, GH, BK, 1536, HH);
    gru_gate_kernel<<<128, 256, 0, stream>>>(GI, GH, last_hidden, HNEW);

    // --- q = h_new @ w1 ; qb = h_new . b1
    int T_q = (BK / 16) * (HH / 16);                // 128 tiles
    gemm_wmma_f32<0><<<(T_q + 7) / 8, 256, 0, stream>>>(HNEW, w1, nullptr, Q, BK, HH, HH);
    dot_hb1_kernel<<<BK, 512, 0, stream>>>(HNEW, b1, QB);

    // --- streaming attention scores over encoder_outputs (HBM-bound pass)
    attn_score_kernel<<<dim3(BK, 8), 512, 0, stream>>>(enc, Q, QB, mask_in, ATTN);

    // --- softmax + top-4
    softmax_topk_kernel<<<BK, 1024, 0, stream>>>(ATTN, ALIGN, SC, SE);

    // --- raw context accumulation (second pass, L2-resident)
    ctx_accum_kernel<<<BK, 512, 0, stream>>>(enc, ALIGN, CTXR);

    // --- context = ctx_raw @ w2^T + b2
    gemm_wmma_f32<1><<<(T_q + 7) / 8, 256, 0, stream>>>(CTXR, w2, b2, CTX, BK, HH, HH);

    // --- result = [context, h_new] @ w3^T + b3
    concat_kernel<<<256, 256, 0, stream>>>(CTX, HNEW, CAT);
    gemm_wmma_f32<1><<<(T_q + 7) / 8, 256, 0, stream>>>(CAT, w3, b3, RES, BK, HH, 1024);

    // --- beam reselection + outputs
    select_kernel<<<1, 32, 0, stream>>>(SC, SE, ev_sc, ev_idx, SARR, SEL, out_scores, out_idx);
    gather_rh_kernel<<<128, 256, 0, stream>>>(RES, HNEW, SARR, out_result, out_hidden);
    gather_attn_kernel<<<1024, 256, 0, stream>>>(attn_prev, ATTN, SARR, out_attn);
    gather_mask_kernel<<<256, 256, 0, stream>>>(mask_in, SARR, SEL, out_mask);
}